// DAMWrapper_53257594471022
// MI455X (gfx1250) — compile-verified
//
#include <hip/hip_runtime.h>
#include <cstdint>
#include <cstddef>

// ---------------- problem constants (from reference) ----------------
constexpr int   kH   = 16;
constexpr int   kN   = 2048;
constexpr float kEPS = 1e-5f;

constexpr int TPB  = 256;  // 8 wave32 per block
constexpr int ROWS = 8;    // output rows per block in the broadcast kernel

typedef float f32x4 __attribute__((ext_vector_type(4)));
typedef int   vint4 __attribute__((vector_size(16)));   // matches builtin param

typedef __attribute__((address_space(1))) vint4* gbl_v4i_ptr;
typedef __attribute__((address_space(3))) vint4* lds_v4i_ptr;

// Probe for the gfx1250 async global->LDS builtin; otherwise use raw ISA asm.
#if defined(__has_builtin)
#if __has_builtin(__builtin_amdgcn_global_load_async_to_lds_b128)
#define ASYNC_LDS_BUILTIN 1
#endif
#endif

__device__ __forceinline__ int iabs_(int x) { return x < 0 ? -x : x; }

// ---------------------------------------------------------------
// Kernel 1: m[h*N+d] = softmax over last dim (2 elems), component 0
//   l_k = alpha_k - log(noise_k + eps);  m = 1/(1+exp(l1-l0))
// H*N = 32768 elements -> 128 blocks of 256. Trivial cost.
// ---------------------------------------------------------------
__global__ void __launch_bounds__(TPB)
gumbel_m_kernel(const float* __restrict__ alphas,
                const float* __restrict__ noise,
                float* __restrict__ m) {
  const int idx = blockIdx.x * TPB + threadIdx.x;          // 0 .. H*N-1 exactly
  const float2 a = reinterpret_cast<const float2*>(alphas)[idx];
  const float2 e = reinterpret_cast<const float2*>(noise)[idx];
  const float l0 = a.x - __logf(e.x + kEPS);
  const float l1 = a.y - __logf(e.y + kEPS);
  m[idx] = 1.0f / (1.0f + __expf(l1 - l0));
}

// ---------------------------------------------------------------
// Kernel 2: masks[h,i,j] = m[h,|i-j|]; norm = 10000*masks - 10000
// Block = (h, 8 rows of i). Stage m[h,:] (8 KB) in LDS via async
// b128 copies, then stream NT b128 stores (store-BW bound).
// ---------------------------------------------------------------
__global__ void __launch_bounds__(TPB)
mask_broadcast_kernel(const float* __restrict__ m,
                      float* __restrict__ out_norm,
                      float* __restrict__ out_mask) {
  __shared__ float sm[kN];                                  // 8 KB
  const int t = threadIdx.x;
  const int h = blockIdx.y;
  const float* mrow = m + (size_t)h * kN;

  // ---- stage m row into LDS: 2048 floats = 512 lane-b128s = 2 per thread ----
#if ASYNC_LDS_BUILTIN
  __builtin_amdgcn_global_load_async_to_lds_b128(
      (gbl_v4i_ptr)(mrow + 4 * t),
      (lds_v4i_ptr)(&sm[4 * t]), 0, 0);
  __builtin_amdgcn_global_load_async_to_lds_b128(
      (gbl_v4i_ptr)(mrow + 1024 + 4 * t),
      (lds_v4i_ptr)(&sm[1024 + 4 * t]), 0, 0);
  asm volatile("s_wait_asynccnt 0" ::: "memory");
#else
  {
    // Raw ISA path: GVS mode (SADDR = 64-bit base, VADDR = 32-bit byte offset),
    // VDST = LDS byte address (low 32 bits of flat shared pointer).
    uint32_t lds0 = (uint32_t)(uintptr_t)(&sm[4 * t]);
    uint32_t lds1 = (uint32_t)(uintptr_t)(&sm[1024 + 4 * t]);
    uint32_t off0 = 16u * (uint32_t)t;
    uint32_t off1 = 4096u + 16u * (uint32_t)t;
    asm volatile(
        "global_load_async_to_lds_b128 %0, %2, %4 offset:0\n\t"
        "global_load_async_to_lds_b128 %1, %3, %4 offset:0\n\t"
        "s_wait_asynccnt 0"
        :
        : "v"(lds0), "v"(lds1), "v"(off0), "v"(off1), "s"(mrow)
        : "memory");
  }
#endif
  __syncthreads();

  // ---- stream the two 2048-float output rows per i, NT b128 stores ----
  const int i0 = blockIdx.x * ROWS;
#pragma unroll
  for (int r = 0; r < ROWS; ++r) {
    const int i = i0 + r;
    const size_t rowbase = (size_t)(h * kN + i) * (size_t)kN;
    f32x4* __restrict__ o0 = reinterpret_cast<f32x4*>(out_norm + rowbase);
    f32x4* __restrict__ o1 = reinterpret_cast<f32x4*>(out_mask + rowbase);
#pragma unroll
    for (int half = 0; half < 2; ++half) {
      const int q = t + half * TPB;                         // float4 index in row
      const int j = 4 * q;
      f32x4 v;
      v.x = sm[iabs_(i - j)];
      v.y = sm[iabs_(i - j - 1)];
      v.z = sm[iabs_(i - j - 2)];
      v.w = sm[iabs_(i - j - 3)];
      f32x4 nv;
      nv.x = fmaf(v.x, 1.0e4f, -1.0e4f);                    // (1-v)*-1e4
      nv.y = fmaf(v.y, 1.0e4f, -1.0e4f);
      nv.z = fmaf(v.z, 1.0e4f, -1.0e4f);
      nv.w = fmaf(v.w, 1.0e4f, -1.0e4f);
      __builtin_nontemporal_store(v, &o1[q]);
      __builtin_nontemporal_store(nv, &o0[q]);
    }
  }
}

// ---------------------------------------------------------------
extern "C" void kernel_launch(void* const* d_in, const int* in_sizes, int n_in,
                              void* d_out, int out_size, void* d_ws, size_t ws_size,
                              hipStream_t stream) {
  (void)in_sizes; (void)n_in; (void)out_size; (void)ws_size;
  const float* alphas = (const float*)d_in[0];   // (H, N, 2) f32
  const float* noise  = (const float*)d_in[1];   // (H, N, 2) f32
  float* mbuf = (float*)d_ws;                    // H*N floats = 128 KB scratch

  float* out_norm = (float*)d_out;                               // (H,N,N)
  float* out_mask = out_norm + (size_t)kH * kN * kN;             // (H,N,N)

  gumbel_m_kernel<<<(kH * kN) / TPB, TPB, 0, stream>>>(alphas, noise, mbuf);

  dim3 grid(kN / ROWS, kH);                      // (256, 16) blocks
  mask_broadcast_kernel<<<grid, TPB, 0, stream>>>(mbuf, out_norm, out_mask);
}